// SA_GAT_30442728194682
// MI455X (gfx1250) — compile-verified
//
#include <hip/hip_runtime.h>
#include <math.h>

typedef float f32x2 __attribute__((ext_vector_type(2)));
typedef float f32x4 __attribute__((ext_vector_type(4)));
typedef float f32x8 __attribute__((ext_vector_type(8)));

#define GS   7     // 7x7 node grid
#define NN   49    // nodes
#define NE   361   // edges
#define NH   8     // heads
#define NB   64    // batch
#define NC   256   // channels
#define HW   56    // spatial
#define BLOCK 256
#define WGPB 4     // workgroups per batch
#define CPW  64    // channels per workgroup
#define ROWQ 14    // float4 per 56-float row
#define PSLOTS (HW*ROWQ)   // 784 float4-slots per channel plane
#define CH_STRIDE4 784     // 56*56/4 float4 per channel

extern "C" __global__ void __launch_bounds__(64)
sa_gat_zero_flags(int* __restrict__ flags) {
    if (threadIdx.x < NB) flags[threadIdx.x] = 0;
}

extern "C" __global__ void __launch_bounds__(BLOCK)
sa_gat_fused(const float* __restrict__ in, const float* __restrict__ Wl,
             const float* __restrict__ asrc, const float* __restrict__ adst,
             const float* __restrict__ Wout, const float* __restrict__ bout,
             const int* __restrict__ eidx, float* __restrict__ out,
             int* __restrict__ flags, float* __restrict__ partials)
{
    __shared__ float nsum[NN];
    __shared__ float xnode[NN];
    __shared__ int   se[NE], de[NE];
    __shared__ float hfeat[NN * NH];
    __shared__ float ebuf[NE * NH];
    __shared__ float aggl[64 * NH];      // padded to 64 rows for WMMA tiles
    __shared__ float scores[64];
    __shared__ float cW[NH], cAs[NH], cAd[NH], cWo[NH], cB[1];

    const int tid   = threadIdx.x;
    const int batch = blockIdx.x >> 2;
    const int wg    = blockIdx.x & 3;
    const size_t slab_off = ((size_t)batch * NC + (size_t)wg * CPW) * (HW * HW);
    const f32x4* in4 = (const f32x4*)(in + slab_off);

    if (tid < NN) nsum[tid] = 0.0f;
    if (tid < NH) { cW[tid] = Wl[tid]; cAs[tid] = asrc[tid]; cAd[tid] = adst[tid]; cWo[tid] = Wout[tid]; }
    if (tid == 0) cB[0] = bout[0];
    __syncthreads();

    // ---------------- Phase 1: pooled partial sums over 64-channel slab ----
    for (int p = tid; p < PSLOTS; p += BLOCK) {
        const int h = p / ROWQ, q = p - h * ROWQ;
        const int node = (h >> 3) * GS + (q >> 1);   // q*4/8 == q/2
        f32x4 acc = {0.f, 0.f, 0.f, 0.f};
        const f32x4* ptr = in4 + p;
        #pragma unroll 8
        for (int c = 0; c < CPW; ++c) acc += ptr[c * CH_STRIDE4];
        atomicAdd(&nsum[node], acc.x + acc.y + acc.z + acc.w);
    }
    __syncthreads();
    if (tid < NN) partials[(size_t)(batch * WGPB + wg) * 64 + tid] = nsum[tid];
    __syncthreads();

    // ---------------- Cross-WG sync within the batch (release/acquire) -----
    if (tid == 0) {
        __hip_atomic_fetch_add(&flags[batch], 1, __ATOMIC_RELEASE, __HIP_MEMORY_SCOPE_AGENT);
        while (__hip_atomic_load(&flags[batch], __ATOMIC_ACQUIRE, __HIP_MEMORY_SCOPE_AGENT) < WGPB)
            __builtin_amdgcn_s_sleep(1);
    }
    __syncthreads();

    if (tid < NN) {
        float s = 0.f;
        #pragma unroll
        for (int w = 0; w < WGPB; ++w)
            s += partials[(size_t)(batch * WGPB + w) * 64 + tid];
        xnode[tid] = s * (1.0f / 16384.0f);   // mean over C*8*8
    }
    for (int t = tid; t < NE; t += BLOCK) { se[t] = eidx[t]; de[t] = eidx[NE + t]; }
    for (int t = tid; t < 64 * NH; t += BLOCK) aggl[t] = 0.0f;
    __syncthreads();

    // ---------------- Phase 2: GAT on the 49-node graph --------------------
    for (int t = tid; t < NN * NH; t += BLOCK) {
        const int n = t >> 3, hd = t & 7;
        hfeat[t] = xnode[n] * cW[hd];
    }
    __syncthreads();

    for (int t = tid; t < NE * NH; t += BLOCK) {
        const int e = t >> 3, hd = t & 7;
        const float v = hfeat[se[e] * NH + hd] * cAs[hd] + hfeat[de[e] * NH + hd] * cAd[hd];
        ebuf[t] = (v > 0.f) ? v : 0.2f * v;   // LeakyReLU(0.2)
    }
    __syncthreads();

    for (int t = tid; t < NN * NH; t += BLOCK) {
        const int n = t >> 3, hd = t & 7;
        float m = -3.4e38f;
        for (int e = 0; e < NE; ++e)
            if (de[e] == n) m = fmaxf(m, ebuf[e * NH + hd]);
        float den = 0.f, ag = 0.f;
        for (int e = 0; e < NE; ++e)
            if (de[e] == n) {
                const float w = __expf(ebuf[e * NH + hd] - m);
                den += w;
                ag  += w * hfeat[se[e] * NH + hd];
            }
        aggl[n * NH + hd] = ag / den;
    }
    __syncthreads();

    // ---------------- scores = sigmoid(agg @ W_out + b) via WMMA -----------
    // Two chained V_WMMA_F32_16X16X4_F32 (K = 8 heads). W_out replicated
    // across all 16 N-columns so every column of D equals agg @ W_out.
    // Uniform path: all 8 waves execute with EXEC == all 1s (ISA requirement);
    // waves 4..7 redundantly recompute tiles 0..3 and skip the write.
    {
        const int lane = tid & 31;
        const int wave = tid >> 5;
        const int tile = wave & 3;
        const int m    = tile * 16 + (lane & 15);
        const int kb   = (lane < 16) ? 0 : 2;     // 16x4 f32 A layout: lanes 0-15 K={0,1}, 16-31 K={2,3}
        f32x2 A0, A1, B0, B1;
        A0.x = aggl[m * NH + kb];     A0.y = aggl[m * NH + kb + 1];
        A1.x = aggl[m * NH + kb + 4]; A1.y = aggl[m * NH + kb + 5];
        B0.x = cWo[kb];               B0.y = cWo[kb + 1];
        B1.x = cWo[kb + 4];           B1.y = cWo[kb + 5];
        f32x8 d = {0.f, 0.f, 0.f, 0.f, 0.f, 0.f, 0.f, 0.f};
        d = __builtin_amdgcn_wmma_f32_16x16x4_f32(false, A0, false, B0, (short)0, d, false, false);
        d = __builtin_amdgcn_wmma_f32_16x16x4_f32(false, A1, false, B1, (short)0, d, false, false);
        if (wave < 4 && (lane == 0 || lane == 16)) {
            // VGPR r: lanes 0-15 -> M=r (N=lane), lanes 16-31 -> M=8+r (N=lane-16)
            const int base = tile * 16 + ((lane == 16) ? 8 : 0);
            #pragma unroll
            for (int r = 0; r < 8; ++r)
                scores[base + r] = 1.0f / (1.0f + __expf(-(d[r] + cB[0])));
        }
    }
    __syncthreads();

    // ---------------- Phase 3: rescale slab (L2-hot reload, NT stores) -----
    f32x4* out4 = (f32x4*)(out + slab_off);
    for (int p = tid; p < PSLOTS; p += BLOCK) {
        const int h = p / ROWQ, q = p - h * ROWQ;
        const float sc = scores[(h >> 3) * GS + (q >> 1)];
        const f32x4* ipp = in4 + p;
        f32x4*       opp = out4 + p;
        #pragma unroll 4
        for (int c = 0; c < CPW; ++c) {
            f32x4 v = ipp[c * CH_STRIDE4];
            v = v * sc;
            __builtin_nontemporal_store(v, &opp[c * CH_STRIDE4]);
        }
    }
}

extern "C" void kernel_launch(void* const* d_in, const int* in_sizes, int n_in,
                              void* d_out, int out_size, void* d_ws, size_t ws_size,
                              hipStream_t stream) {
    (void)in_sizes; (void)n_in; (void)out_size; (void)ws_size;
    const float* in    = (const float*)d_in[0];
    const float* Wl    = (const float*)d_in[1];
    const float* a_src = (const float*)d_in[2];
    const float* a_dst = (const float*)d_in[3];
    const float* W_out = (const float*)d_in[4];
    const float* b_out = (const float*)d_in[5];
    const int*   eidx  = (const int*)d_in[6];
    float* out = (float*)d_out;

    int*   flags    = (int*)d_ws;                       // 64 ints
    float* partials = (float*)((char*)d_ws + 256);      // 64 batches * 4 wg * 64 floats

    sa_gat_zero_flags<<<1, 64, 0, stream>>>(flags);
    sa_gat_fused<<<NB * WGPB, BLOCK, 0, stream>>>(in, Wl, a_src, a_dst, W_out, b_out,
                                                  eidx, out, flags, partials);
}